// PeriodicShift_79834852098336
// MI455X (gfx1250) — compile-verified
//
#include <hip/hip_runtime.h>
#include <hip/hip_bf16.h>

typedef float v2f __attribute__((ext_vector_type(2)));
typedef float v8f __attribute__((ext_vector_type(8)));

// ---- constants from the reference ----
#define NHEADS   12
#define NTOK     196
#define CCH      64
#define BTOT     256
#define ROWSTRIDE 150528   /* 12*196*64 floats per Bidx */
#define HSTRIDE   12544    /* 196*64 */
#define TSTRIDE   602112   /* 4*150528 : +1 frame in Bidx space */
#define NPOS     50176     /* 256*196 = positions over both batches */
#define NPOSB    25088     /* positions per batch */

// ------------------------------------------------------------------
// Kernel A: attn[dir][Bidx*196+n] = <x[Bidx,:,n,:], pool_w>   (one wave/pos)
// ------------------------------------------------------------------
__global__ void attn_kernel(const float* __restrict__ x,
                            const float* __restrict__ pwF,
                            const float* __restrict__ pwB,
                            float* __restrict__ attn) {
  int wid  = threadIdx.x >> 5;
  int lane = threadIdx.x & 31;
  int pos  = blockIdx.x * 8 + wid;
  if (pos >= NPOS) return;
  int Bi = pos / NTOK, n = pos - Bi * NTOK;
  const float* base = x + (size_t)Bi * ROWSTRIDE + (size_t)n * CCH;
  float sF = 0.f, sB = 0.f;
#pragma unroll
  for (int i = 0; i < 24; ++i) {
    int d = lane + (i << 5);
    int h = d >> 6, c = d & 63;
    float v = base[(size_t)h * HSTRIDE + c];
    sF += v * pwF[d];
    sB += v * pwB[d];
  }
#pragma unroll
  for (int off = 16; off; off >>= 1) {
    sF += __shfl_xor(sF, off, 32);
    sB += __shfl_xor(sB, off, 32);
  }
  if (lane == 0) { attn[pos] = sF; attn[NPOS + pos] = sB; }
}

// ------------------------------------------------------------------
// Kernel B: exact entmax-1.5 over 25088 values via tau-bisection.
// One block per (dir,b). Overwrites scores with weights in place.
// ------------------------------------------------------------------
__global__ void entmax_big_kernel(float* __restrict__ attn) {
  __shared__ float red[1024];
  int dir = blockIdx.x >> 1, b = blockIdx.x & 1;
  float* p = attn + (size_t)dir * NPOS + (size_t)b * NPOSB;
  int tid = threadIdx.x;

  float m = -1e30f;
  for (int i = tid; i < NPOSB; i += 1024) m = fmaxf(m, p[i]);
  red[tid] = m; __syncthreads();
  for (int s = 512; s > 0; s >>= 1) {
    if (tid < s) red[tid] = fmaxf(red[tid], red[tid + s]);
    __syncthreads();
  }
  float mx = red[0]; __syncthreads();

  float lo = -1.0f, hi = 0.0f;
  for (int it = 0; it < 30; ++it) {
    float tau = 0.5f * (lo + hi);
    float s = 0.f;
    for (int i = tid; i < NPOSB; i += 1024) {
      float z = fmaxf((p[i] - mx) * 0.5f - tau, 0.f);
      s += z * z;
    }
    red[tid] = s; __syncthreads();
    for (int k = 512; k > 0; k >>= 1) {
      if (tid < k) red[tid] += red[tid + k];
      __syncthreads();
    }
    float S = red[0]; __syncthreads();
    if (S > 1.0f) lo = tau; else hi = tau;
  }
  float tau = 0.5f * (lo + hi);
  for (int i = tid; i < NPOSB; i += 1024) {
    float z = fmaxf((p[i] - mx) * 0.5f - tau, 0.f);
    p[i] = z * z;
  }
}

// ------------------------------------------------------------------
// Kernel C: pooled[dir][b][768] += w * x  (fused fwd+bwd, single x pass)
// grid = Bidx*h = 3072 blocks, 64 threads (c)
// ------------------------------------------------------------------
__global__ void pool_kernel(const float* __restrict__ x,
                            const float* __restrict__ w,
                            float* __restrict__ pooled) {
  int h  = blockIdx.x % NHEADS;
  int Bi = blockIdx.x / NHEADS;
  int c  = threadIdx.x;
  int b  = Bi >> 7;
  const float* wF = w + (size_t)Bi * NTOK;
  const float* wB = wF + NPOS;
  const float* base = x + (size_t)Bi * ROWSTRIDE + (size_t)h * HSTRIDE + c;
  float aF = 0.f, aB = 0.f;
  for (int n = 0; n < NTOK; ++n) {
    float v = base[n * CCH];
    aF += wF[n] * v;
    aB += wB[n] * v;
  }
  atomicAdd(&pooled[b * 768 + h * CCH + c], aF);
  atomicAdd(&pooled[1536 + b * 768 + h * CCH + c], aB);
}

// ------------------------------------------------------------------
// Kernel D: win = pooled@win_w+win_b -> entmax15 per (b,f) over 11 taps,
// then build banded T[b][dir][f][t_in][t_out] (scaled by 1/8 mean-fold).
// grid = 4 blocks (dir,b), 128 threads
// ------------------------------------------------------------------
__global__ void win_kernel(const float* __restrict__ pooled,
                           const float* __restrict__ fww, const float* __restrict__ fwb,
                           const float* __restrict__ bww, const float* __restrict__ bwb,
                           float* __restrict__ Tmat) {
  __shared__ float ps[768];
  __shared__ float wv[88];
  __shared__ float wsh[88];   // entmax'd windows [f][11], pre-scaled by 1/8
  int dir = blockIdx.x >> 1, b = blockIdx.x & 1;
  int tid = threadIdx.x;
  const float* ww = dir ? bww : fww;
  const float* wb = dir ? bwb : fwb;

  for (int i = tid; i < 768; i += 128) ps[i] = pooled[dir * 1536 + b * 768 + i];
  __syncthreads();

  if (tid < 88) {
    float s = wb[tid];
    for (int d = 0; d < 768; ++d) s += ps[d] * ww[(size_t)d * 88 + tid];
    wv[tid] = s;
  }
  __syncthreads();

  if (tid < 8) {
    int f = tid;
    float z[11], xs[11], taus[11];
    float mx = -1e30f;
    for (int j = 0; j < 11; ++j) { z[j] = wv[j * 8 + f]; mx = fmaxf(mx, z[j]); }
    for (int j = 0; j < 11; ++j) { z[j] = (z[j] - mx) * 0.5f; xs[j] = z[j]; }
    for (int a = 1; a < 11; ++a) {            // insertion sort, descending
      float key = xs[a]; int q = a - 1;
      while (q >= 0 && xs[q] < key) { xs[q + 1] = xs[q]; --q; }
      xs[q + 1] = key;
    }
    float cum = 0.f, cum2 = 0.f;
    for (int r = 1; r <= 11; ++r) {
      float v = xs[r - 1]; cum += v; cum2 += v * v;
      float mean = cum / r, ms = cum2 / r;
      float ss = r * (ms - mean * mean);
      float delta = (1.0f - ss) / r;
      float sq = (delta > 0.f) ? sqrtf(delta) : 0.f;
      taus[r - 1] = mean - sq;
    }
    int support = 0;
    for (int r = 0; r < 11; ++r) support += (taus[r] <= xs[r]) ? 1 : 0;
    float tau = taus[support - 1];
    for (int j = 0; j < 11; ++j) {
      float y = fmaxf(z[j] - tau, 0.f);
      wsh[f * 11 + j] = y * y * 0.125f;   // fold mean over f
    }
  }
  __syncthreads();

  float* Tb = Tmat + (size_t)(b * 2 + dir) * 8192;
  for (int idx = tid; idx < 8192; idx += 128) {
    int f = idx >> 10, rem = idx & 1023;
    int ti = rem >> 5, to = rem & 31;
    float val = 0.f;
    if (dir == 0) {                 // fwd: t_out - t_in in [10,20], j = 20-(to-ti)
      int dlt = to - ti;
      if (dlt >= 10 && dlt <= 20) val = wsh[f * 11 + (20 - dlt)];
    } else {                        // bwd: t_in - t_out in [10,20], j = dlt-10
      int dlt = ti - to;
      if (dlt >= 10 && dlt <= 20) val = wsh[f * 11 + (dlt - 10)];
    }
    Tb[idx] = val;
  }
}

// ------------------------------------------------------------------
// Kernel E: temporal-mixing matmul via V_WMMA_F32_16X16X4_F32.
// Per wave: 16 m-rows (m = n*8+cp), N=32 t_out (2 halves), K=32 t_in,
// f-loop of 8, fwd+bwd accumulators share the A loads.
// LDS holds T[b] (2 dir x 8 f x 32x32 = 64KB, dynamic).
// grid = 2352 blocks x 128 threads (4 waves)
// ------------------------------------------------------------------
__global__ void conv_wmma_kernel(const float* __restrict__ x,
                                 const float* __restrict__ Tmat,
                                 float* __restrict__ out) {
  extern __shared__ float Ts[];   // 16384 floats
  int tid = threadIdx.x;
  int gw  = blockIdx.x * 4 + (tid >> 5);
  int b   = gw / 4704;            // uniform within a block (4704 = 1176*4)
  for (int i = tid; i < 16384; i += 128) Ts[i] = Tmat[(size_t)b * 16384 + i];
  __syncthreads();

  int r  = gw - b * 4704;
  int k  = r / 1176;  int r2 = r - k * 1176;
  int h  = r2 / 98;   int mt = r2 - h * 98;
  int lane   = tid & 31;
  int laneM  = lane & 15;
  int laneHi = lane >> 4;

  int m  = mt * 16 + laneM;
  int nA = m >> 3, cpA = m & 7;
  const float* Ab = x + (size_t)(b * 128 + k) * ROWSTRIDE
                      + (size_t)h * HSTRIDE + (size_t)nA * CCH + cpA * 8;

  v8f aF0 = {0.f,0.f,0.f,0.f,0.f,0.f,0.f,0.f};
  v8f aF1 = aF0, aB0 = aF0, aB1 = aF0;

  for (int f = 0; f < 8; ++f) {
    const float* Af = Ab + f;
#pragma unroll
    for (int kc = 0; kc < 8; ++kc) {
      int t0 = kc * 4 + laneHi * 2;
      v2f a;
      a.x = Af[(size_t)t0 * TSTRIDE];
      a.y = Af[(size_t)(t0 + 1) * TSTRIDE];

      const float* Trow = Ts + f * 1024 + t0 * 32;        // dir 0 (fwd)
      v2f b0, b1;
      b0.x = Trow[laneM];      b0.y = Trow[32 + laneM];
      b1.x = Trow[16 + laneM]; b1.y = Trow[48 + laneM];
      const float* Trw2 = Trow + 8192;                    // dir 1 (bwd)
      v2f c0, c1;
      c0.x = Trw2[laneM];      c0.y = Trw2[32 + laneM];
      c1.x = Trw2[16 + laneM]; c1.y = Trw2[48 + laneM];

      aF0 = __builtin_amdgcn_wmma_f32_16x16x4_f32(false, a, false, b0, (short)0, aF0, false, false);
      aF1 = __builtin_amdgcn_wmma_f32_16x16x4_f32(false, a, false, b1, (short)0, aF1, false, false);
      aB0 = __builtin_amdgcn_wmma_f32_16x16x4_f32(false, a, false, c0, (short)0, aB0, false, false);
      aB1 = __builtin_amdgcn_wmma_f32_16x16x4_f32(false, a, false, c1, (short)0, aB1, false, false);
    }
  }

  // Writeout: lane holds, per N-half, channels 0..7 (fwd rows) + 8..15 (bwd)
  // at (n = mt*2 + laneHi, t_out = nh*16 + laneM) -> 16 contiguous floats.
  int nn = mt * 2 + laneHi;
#pragma unroll
  for (int nh = 0; nh < 2; ++nh) {
    int to = nh * 16 + laneM;
    float* ob = out + (size_t)(b * 128 + to * 4 + k) * ROWSTRIDE
                    + (size_t)h * HSTRIDE + (size_t)nn * CCH;
    v8f F = nh ? aF1 : aF0;
    v8f Bv = nh ? aB1 : aB0;
    *(float4*)(ob +  0) = make_float4(F[0], F[1], F[2], F[3]);
    *(float4*)(ob +  4) = make_float4(F[4], F[5], F[6], F[7]);
    *(float4*)(ob +  8) = make_float4(Bv[0], Bv[1], Bv[2], Bv[3]);
    *(float4*)(ob + 12) = make_float4(Bv[4], Bv[5], Bv[6], Bv[7]);
  }
}

// ------------------------------------------------------------------
// Kernel F: passthrough channels 16..63 (coalesced float4 copy)
// ------------------------------------------------------------------
__global__ void copy_tail_kernel(const float* __restrict__ x, float* __restrict__ out) {
  int idx = blockIdx.x * 256 + threadIdx.x;   // < 602112*12
  int row = idx / 12;
  int q   = idx - row * 12;
  size_t off = (size_t)row * CCH + 16 + q * 4;
  *(float4*)(out + off) = *(const float4*)(x + off);
}

extern "C" void kernel_launch(void* const* d_in, const int* in_sizes, int n_in,
                              void* d_out, int out_size, void* d_ws, size_t ws_size,
                              hipStream_t stream) {
  const float* x   = (const float*)d_in[0];
  const float* fpw = (const float*)d_in[3];
  const float* bpw = (const float*)d_in[5];
  const float* fww = (const float*)d_in[7];
  const float* fwb = (const float*)d_in[8];
  const float* bww = (const float*)d_in[9];
  const float* bwb = (const float*)d_in[10];
  float* out = (float*)d_out;

  float* wsf    = (float*)d_ws;
  float* attn   = wsf;                 // 2 * 50176          = 100352 floats
  float* pooled = wsf + 100352;        // 2 * 2 * 768        =   3072 floats
  float* Tmat   = wsf + 103424;        // 2 * 2 * 8 * 1024   =  32768 floats

  attn_kernel<<<6272, 256, 0, stream>>>(x, fpw, bpw, attn);
  entmax_big_kernel<<<4, 1024, 0, stream>>>(attn);
  hipMemsetAsync(pooled, 0, 3072 * sizeof(float), stream);
  pool_kernel<<<3072, 64, 0, stream>>>(x, attn, pooled);
  win_kernel<<<4, 128, 0, stream>>>(pooled, fww, fwb, bww, bwb, Tmat);
  conv_wmma_kernel<<<2352, 128, 65536, stream>>>(x, Tmat, out);
  copy_tail_kernel<<<28224, 256, 0, stream>>>(x, out);
}